// Hole_62405874811502
// MI455X (gfx1250) — compile-verified
//
#include <hip/hip_runtime.h>

typedef __attribute__((ext_vector_type(2))) float v2f;
typedef __attribute__((ext_vector_type(8))) float v8f;

#define DIMS   128
#define BATCH  16384
#define MARGIN 1.0f
#define WPB    8     // waves per block

// HolE score for one (h, r, t) triple, one wave, f32 WMMA.
//   corr[k] = sum_j h[j] * t[(j+k) mod 128]
//   score   = sum_k r[k] * corr[k]
// GEMM form: A[m,j] = h[(j-16m)&127] (16x128), B[j,n] = t[(j+n)&127] (128x16)
//   => D[m,n] = corr[(16m+n) mod 128]; rows m and m+8 of D are duplicates.
// LDS replication (h x3, t x2) removes the &127 wrap so operand fetches are
// pure base+immediate ds loads.
__device__ __forceinline__ float wave_score(const float* __restrict__ hrow,
                                            const float* __restrict__ trow,
                                            float* __restrict__ hl,   // 384 f32: h replicated 3x
                                            float* __restrict__ tl,   // 256 f32: t replicated 2x
                                            const float* __restrict__ rl,
                                            int lane) {
  const float4 hv = *(const float4*)(hrow + lane * 4);
  const float4 tv = *(const float4*)(trow + lane * 4);
  *(float4*)(hl + lane * 4)       = hv;
  *(float4*)(hl + lane * 4 + 128) = hv;
  *(float4*)(hl + lane * 4 + 256) = hv;
  *(float4*)(tl + lane * 4)       = tv;
  *(float4*)(tl + lane * 4 + 128) = tv;
  // LDS ops within a wave complete in order; just block compiler reordering.
  __asm volatile("" ::: "memory");

  const int half = lane >> 4;   // 0 -> K{0,1}, 1 -> K{2,3} of each 4-wide chunk
  const int mn   = lane & 15;   // M for the A fragment, N for the B fragment
  const int d0   = half * 2;

  // Per-lane constant bases; in-loop addresses are base + 16*s bytes.
  // A index: 256 + d0 - 16*mn + 4s   in [16, 383]  (8B-aligned pair -> ds_load_b64)
  // B index: d0 + mn + 4s            in [0, 143]   (pair -> ds_load_2addr_b32)
  const float* pa = hl + (256 + d0 - 16 * mn);
  const float* pb = tl + (d0 + mn);

  v8f c0 = {}, c1 = {};   // two accumulators to break the WMMA dep chain
  #pragma unroll
  for (int s = 0; s < 32; s += 2) {
    v2f a0 = *(const v2f*)(pa + 4 * s);
    v2f b0; b0.x = pb[4 * s];     b0.y = pb[4 * s + 1];
    c0 = __builtin_amdgcn_wmma_f32_16x16x4_f32(false, a0, false, b0,
                                               (short)0, c0, false, false);
    v2f a1 = *(const v2f*)(pa + 4 * s + 4);
    v2f b1; b1.x = pb[4 * s + 4]; b1.y = pb[4 * s + 5];
    c1 = __builtin_amdgcn_wmma_f32_16x16x4_f32(false, a1, false, b1,
                                               (short)0, c1, false, false);
  }

  // C/D layout: lane<16, VGPR v -> D[v, lane]; lane>=16, VGPR v -> D[v+8, lane-16].
  // D[v+8,n] == D[v,n], so every lane dots its copy with r[16v+n]; the 32-lane
  // sum counts the score exactly twice -> scale by 0.5.
  float part = 0.0f;
  #pragma unroll
  for (int v = 0; v < 8; ++v)
    part += rl[16 * v + mn] * (c0[v] + c1[v]);
  #pragma unroll
  for (int off = 16; off > 0; off >>= 1)
    part += __shfl_xor(part, off, 32);
  return part * 0.5f;
}

__global__ __launch_bounds__(256)
void hole_score_kernel(const float* __restrict__ ent,
                       const float* __restrict__ rel,
                       const int*   __restrict__ data,
                       float*       __restrict__ per_row) {
  __shared__ __align__(16) float hls[WPB][384];
  __shared__ __align__(16) float tls[WPB][256];
  __shared__ __align__(16) float rls[WPB][DIMS];

  const int wid  = threadIdx.x >> 5;
  const int lane = threadIdx.x & 31;
  const int row  = blockIdx.x * WPB + wid;

  const int* d = data + (size_t)row * 5;
  const size_t hI  = (size_t)d[0];
  const size_t tI  = (size_t)d[1];
  const size_t rI  = (size_t)d[2];
  const size_t chI = (size_t)d[3];
  const size_t ctI = (size_t)d[4];

  const float4 rv = *(const float4*)(rel + rI * DIMS + lane * 4);
  *(float4*)(&rls[wid][lane * 4]) = rv;
  __asm volatile("" ::: "memory");

  // Warm caches for the negative-sample rows while the positive score runs.
  __builtin_prefetch(ent + chI * DIMS + lane * 4, 0, 0);
  __builtin_prefetch(ent + ctI * DIMS + lane * 4, 0, 0);

  const float pos = wave_score(ent + hI * DIMS, ent + tI * DIMS,
                               hls[wid], tls[wid], rls[wid], lane);
  const float neg = wave_score(ent + chI * DIMS, ent + ctI * DIMS,
                               hls[wid], tls[wid], rls[wid], lane);

  if (lane == 0) {
    const float v = pos - neg + MARGIN;
    per_row[row] = v > 0.0f ? v : 0.0f;
  }
}

// Deterministic mean over BATCH values: fixed per-thread serial order + LDS tree.
__global__ __launch_bounds__(256)
void hole_reduce_kernel(const float* __restrict__ per_row, float* __restrict__ out) {
  __shared__ float buf[256];
  const int t = threadIdx.x;
  float s = 0.0f;
  const int chunk = BATCH / 256;  // 64
  #pragma unroll 8
  for (int i = 0; i < chunk; ++i)
    s += per_row[t * chunk + i];
  buf[t] = s;
  __syncthreads();
  for (int off = 128; off > 0; off >>= 1) {
    if (t < off) buf[t] += buf[t + off];
    __syncthreads();
  }
  if (t == 0) out[0] = buf[0] * (1.0f / (float)BATCH);
}

extern "C" void kernel_launch(void* const* d_in, const int* in_sizes, int n_in,
                              void* d_out, int out_size, void* d_ws, size_t ws_size,
                              hipStream_t stream) {
  const float* ent  = (const float*)d_in[0];   // 1,000,000 x 128 f32
  const float* rel  = (const float*)d_in[1];   // 1,000 x 128 f32
  const int*   data = (const int*)d_in[2];     // 16384 x 5 int32
  float* per_row = (float*)d_ws;               // 16384 f32 scratch
  float* out     = (float*)d_out;              // 1 f32

  hole_score_kernel<<<BATCH / WPB, 256, 0, stream>>>(ent, rel, data, per_row);
  hole_reduce_kernel<<<1, 256, 0, stream>>>(per_row, out);
}